// PlenoxelModel_18880676233217
// MI455X (gfx1250) — compile-verified
//
#include <hip/hip_runtime.h>
#include <math.h>

typedef __attribute__((ext_vector_type(2))) float v2f;
typedef __attribute__((ext_vector_type(8))) float v8f;

#define GRD  128
#define SAMP 128
#define VD   28

// SH matrix entry A[m][j]: channel row m in 0..2, flat K index j = c*9+n (27 used, padded to 28).
// j is a compile-time constant after unrolling; m is runtime (lane&15).
#define SHA(jj) ((((jj) < 27) && (m == ((jj) / 9))) ? Ysh[(jj) % 9] : 0.0f)
// Coefficient with zero padding for j >= 27 (constant index after unroll).
#define CFX(jj) (((jj) < 27) ? cf[((jj) < 27) ? (jj) : 0] : 0.0f)

__global__ __launch_bounds__(128) void plenoxel_render_kernel(
    const float* __restrict__ grid,   // [128,128,128,28]
    const float* __restrict__ spos,   // [R,128,3]
    const float* __restrict__ sdist,  // [R,128]
    const float* __restrict__ vang,   // [R,2]
    float* __restrict__ out)          // [R,3]
{
  const int r    = blockIdx.x;
  const int s    = threadIdx.x;      // sample index within ray, 0..127
  const int lane = s & 31;
  const int wv   = s >> 5;           // wave id 0..3
  const int m    = lane & 15;
  const bool hi  = lane >= 16;

  __shared__ float s_wsum[4];
  __shared__ float s_red[4][3];

  // ---------------- SH basis (per-ray constants, computed in every lane) -------------
  const float th = vang[2 * r + 0];
  const float ph = vang[2 * r + 1];
  const float st = __sinf(th), ct = __cosf(th);
  const float sp = __sinf(ph), cp = __cosf(ph);
  float Ysh[9];
  Ysh[0] = 0.28209479177387814f;
  Ysh[1] = 0.4886025119029199f * st * sp;
  Ysh[2] = 0.4886025119029199f * ct;
  Ysh[3] = 0.4886025119029199f * st * cp;
  Ysh[4] = 1.0925484305920792f * (st * cp) * (st * sp);
  Ysh[5] = 1.0925484305920792f * (st * sp) * ct;
  Ysh[6] = 0.31539156525252005f * (3.0f * ct * ct - 1.0f);
  Ysh[7] = 1.0925484305920792f * (st * cp) * ct;
  Ysh[8] = 0.5462742152960396f * ((st * cp) * (st * cp) - (st * sp) * (st * sp));

  // ---------------- Per-sample position / trilinear setup ----------------------------
  const int rs = r * SAMP + s;
  const float px = spos[3 * rs + 0];
  const float py = spos[3 * rs + 1];
  const float pz = spos[3 * rs + 2];
  const float dd = sdist[rs];

  const int fx0 = (int)floorf(px);
  const int fy0 = (int)floorf(py);
  const int fz0 = (int)floorf(pz);
  const float fx = px - (float)fx0;
  const float fy = py - (float)fy0;
  const float fz = pz - (float)fz0;

  const int cx[2] = { min(max(fx0, 0), GRD - 1), min(max(fx0 + 1, 0), GRD - 1) };
  const int cy[2] = { min(max(fy0, 0), GRD - 1), min(max(fy0 + 1, 0), GRD - 1) };
  const int cz[2] = { min(max(fz0, 0), GRD - 1), min(max(fz0 + 1, 0), GRD - 1) };
  const float wx[2] = { 1.0f - fx, fx };
  const float wy[2] = { 1.0f - fy, fy };
  const float wz[2] = { 1.0f - fz, fz };

  // Corner base pointers (each voxel = 28 floats = 112 B, 16B-aligned) + prefetch.
  const float* bases[8];
#pragma unroll
  for (int c8 = 0; c8 < 8; ++c8) {
    const int dx = (c8 >> 2) & 1, dy = (c8 >> 1) & 1, dz = c8 & 1;
    const int idx = ((cx[dx] * GRD + cy[dy]) * GRD + cz[dz]) * VD;
    bases[c8] = grid + idx;
    __builtin_prefetch(bases[c8], 0, 1);   // global_prefetch_b8: warm both cachelines' head
  }

  // ---------------- Gather 8 corners, accumulate sigma + 27 coeffs -------------------
  float sigma = 0.0f;
  float cf[27];
#pragma unroll
  for (int j = 0; j < 27; ++j) cf[j] = 0.0f;

#pragma unroll
  for (int c8 = 0; c8 < 8; ++c8) {
    const int dx = (c8 >> 2) & 1, dy = (c8 >> 1) & 1, dz = c8 & 1;
    const float w = wx[dx] * wy[dy] * wz[dz];
    const float4* g4 = reinterpret_cast<const float4*>(bases[c8]);
    float fe[28];
#pragma unroll
    for (int q = 0; q < 7; ++q) {          // 7x global_load_b128 per corner
      const float4 v = g4[q];
      fe[4 * q + 0] = v.x; fe[4 * q + 1] = v.y;
      fe[4 * q + 2] = v.z; fe[4 * q + 3] = v.w;
    }
    sigma += w * fe[0];
#pragma unroll
    for (int j = 0; j < 27; ++j) cf[j] += w * fe[j + 1];
  }

  // ---------------- Attenuation + inclusive cumsum over the ray ----------------------
  const float att = __expf(-sigma * dd);

  float incl = att;                        // wave32 inclusive scan (shfl_up ladder)
#pragma unroll
  for (int o = 1; o < 32; o <<= 1) {
    const float n = __shfl_up(incl, o);
    if (lane >= o) incl += n;
  }
  if (lane == 31) s_wsum[wv] = incl;
  __syncthreads();
  float carry = 0.0f;
#pragma unroll
  for (int w2 = 0; w2 < 4; ++w2) carry += (w2 < wv) ? s_wsum[w2] : 0.0f;
  const float trans  = carry + incl;       // inclusive cumsum, matches jnp.cumsum
  const float weight = trans * (1.0f - att);

  // ---------------- SH projection via v_wmma_f32_16x16x4_f32 -------------------------
  // D = A(16x27pad: rows 0..2 = SH matrix) x B(27pad x 16: coeff columns of 16 samples).
  // A layout (ISA 7.12.2, 32-bit A 16x4): VGPR0 = {K, K+2} by lane-half, rows = lane%16.
  // B mirrors the same K split; sample n lives in lane n (low half) / lane n (high half).
  v8f acc0 = {0.f,0.f,0.f,0.f,0.f,0.f,0.f,0.f};   // samples = wave lanes 0..15
  v8f acc1 = {0.f,0.f,0.f,0.f,0.f,0.f,0.f,0.f};   // samples = wave lanes 16..31
#pragma unroll
  for (int t = 0; t < 7; ++t) {            // K tiles of 4, covers 27 (+1 zero pad)
    const int k0 = 4 * t;
    v2f A;
    A.x = hi ? SHA(k0 + 2) : SHA(k0 + 0);
    A.y = hi ? SHA(k0 + 3) : SHA(k0 + 1);
    const float c0 = CFX(k0 + 0), c1 = CFX(k0 + 1);
    const float c2 = CFX(k0 + 2), c3 = CFX(k0 + 3);
#pragma unroll
    for (int h = 0; h < 2; ++h) {
      const int ssrc = (lane & 15) + 16 * h;          // which lane's sample feeds column n
      const float t0 = __shfl(c0, ssrc);
      const float t1 = __shfl(c1, ssrc);
      const float t2 = __shfl(c2, ssrc);
      const float t3 = __shfl(c3, ssrc);
      v2f B;
      B.x = hi ? t2 : t0;
      B.y = hi ? t3 : t1;
      if (h == 0)
        acc0 = __builtin_amdgcn_wmma_f32_16x16x4_f32(false, A, false, B,
                                                     (short)0, acc0, false, false);
      else
        acc1 = __builtin_amdgcn_wmma_f32_16x16x4_f32(false, A, false, B,
                                                     (short)0, acc1, false, false);
    }
  }

  // Extract rgb for this thread's sample: D[m=c][n] sits in element c, lane n (0..15).
  float pc[3];
#pragma unroll
  for (int c = 0; c < 3; ++c) {
    const float lo  = acc0[c];                       // valid for lanes 0..15 (n = lane)
    const float hiv = __shfl(acc1[c], lane & 15);    // lanes 16..31 fetch column lane-16
    const float rgb = hi ? hiv : lo;
    pc[c] = weight * rgb;
  }

  // ---------------- Reduce over 128 samples -> out[r, 0..2] --------------------------
#pragma unroll
  for (int c = 0; c < 3; ++c) {
#pragma unroll
    for (int o = 16; o > 0; o >>= 1) pc[c] += __shfl_xor(pc[c], o);
  }
  if (lane == 0) {
    s_red[wv][0] = pc[0]; s_red[wv][1] = pc[1]; s_red[wv][2] = pc[2];
  }
  __syncthreads();
  if (s == 0) {
#pragma unroll
    for (int c = 0; c < 3; ++c) {
      out[3 * r + c] = s_red[0][c] + s_red[1][c] + s_red[2][c] + s_red[3][c];
    }
  }
}

extern "C" void kernel_launch(void* const* d_in, const int* in_sizes, int n_in,
                              void* d_out, int out_size, void* d_ws, size_t ws_size,
                              hipStream_t stream) {
  const float* grid  = (const float*)d_in[0];
  const float* spos  = (const float*)d_in[1];
  const float* sdist = (const float*)d_in[2];
  const float* vang  = (const float*)d_in[3];
  float* out = (float*)d_out;
  const int R = in_sizes[3] / 2;     // viewing_angles is [R,2]
  plenoxel_render_kernel<<<dim3(R), dim3(128), 0, stream>>>(grid, spos, sdist, vang, out);
}